// GraphNeuralNetwork_64493228916781
// MI455X (gfx1250) — compile-verified
//
#include <hip/hip_runtime.h>

#define N_NODES 100000
#define N_EDGES 1600000
#define F_IN 64
#define N_HIDDEN 64
#define N_GRAPHS 128
#define BN_EPS 1e-3f

typedef __attribute__((ext_vector_type(2))) float v2f;
typedef __attribute__((ext_vector_type(8))) float v8f;

// ---------------------------------------------------------------- zero scratch
__global__ void gnn_zero_kernel(float* __restrict__ p, long n) {
    long i = (long)blockIdx.x * blockDim.x + threadIdx.x;
    long stride = (long)gridDim.x * blockDim.x;
    for (; i < n; i += stride) p[i] = 0.0f;
}

// ------------------------------------------------- fused GEMM + BatchNorm (WMMA)
// h = BN(x @ W + b), inference stats folded into scale/shift:
//   scale[c] = gamma[c] * rsqrt(var[c]+eps)
//   shift[c] = (b[c]-mean[c])*scale[c] + beta[c]
// Each wave computes a 16x64 tile of h with V_WMMA_F32_16X16X4_F32.
// fp32 A layout (16x4): lanes0-15: v0=K0,v1=K1 ; lanes16-31: v0=K2,v1=K3
//   -> a = x[row][k+2*half .. +1] (one global_load_b64 per lane per k-step)
// B staged in LDS pair-interleaved: sB[p*128 + 2*col] = {W[2p][col], W[2p+1][col]}
//   -> b operand is a single aligned ds_load_b64 (no register shuffles)
__launch_bounds__(128)
__global__ void gnn_gemm_bn_kernel(const float* __restrict__ x,
                                   const float* __restrict__ W,
                                   const float* __restrict__ b,
                                   const float* __restrict__ gamma,
                                   const float* __restrict__ beta,
                                   const float* __restrict__ mean,
                                   const float* __restrict__ var,
                                   float* __restrict__ h) {
    __shared__ float sB[32 * 128];      // 16 KB, pair-interleaved W
    __shared__ float sScale[N_HIDDEN];
    __shared__ float sShift[N_HIDDEN];

    const int tid = threadIdx.x;

    // stage W into LDS, interleaving K-row pairs so (W[2p][c], W[2p+1][c])
    // sit in one 8-byte slot
    #pragma unroll
    for (int i = tid; i < 2048; i += 128) {
        const int p   = i >> 6;          // pair index 0..31  (rows 2p, 2p+1)
        const int col = i & 63;
        float2 w2;
        w2.x = W[(2 * p) * N_HIDDEN + col];
        w2.y = W[(2 * p + 1) * N_HIDDEN + col];
        *(float2*)&sB[p * 128 + 2 * col] = w2;
    }
    if (tid < N_HIDDEN) {
        float sc = gamma[tid] * rsqrtf(var[tid] + BN_EPS);
        sScale[tid] = sc;
        sShift[tid] = (b[tid] - mean[tid]) * sc + beta[tid];
    }
    __syncthreads();

    const int wave = tid >> 5;
    const int lane = tid & 31;
    const int lm   = lane & 15;
    const int half = lane >> 4;

    const long m0 = ((long)blockIdx.x * 4 + wave) * 16;
    if (m0 >= N_NODES) return;           // wave-uniform: EXEC stays all-1s below

    v8f acc[4] = {v8f{}, v8f{}, v8f{}, v8f{}};

    const float* xrow = x + (m0 + lm) * (long)F_IN;

    #pragma unroll
    for (int k = 0; k < F_IN; k += 4) {
        // A operand: contiguous float2 per lane (K = k+2*half, k+2*half+1)
        const float2 av = *(const float2*)(xrow + k + 2 * half);
        v2f a; a.x = av.x; a.y = av.y;

        const int pbase = (k >> 1) + half;           // LDS pair row

        #pragma unroll
        for (int t = 0; t < 4; ++t) {
            const int col = t * 16 + lm;
            const float2 bv = *(const float2*)&sB[pbase * 128 + 2 * col];
            v2f bb; bb.x = bv.x; bb.y = bv.y;
            acc[t] = __builtin_amdgcn_wmma_f32_16x16x4_f32(
                /*neg_a=*/false, a, /*neg_b=*/false, bb,
                /*c_mod=*/(short)0, acc[t],
                /*reuse_a=*/false, /*reuse_b=*/false);
        }
    }

    // Fused BN epilogue + store. C/D layout: VGPR r, lanes0-15 -> row m0+r,
    // lanes16-31 -> row m0+r+8, col = t*16 + lm.
    #pragma unroll
    for (int t = 0; t < 4; ++t) {
        const int col = t * 16 + lm;
        const float sc = sScale[col];
        const float sh = sShift[col];
        #pragma unroll
        for (int r = 0; r < 8; ++r) {
            const long row = m0 + r + 8 * half;
            h[row * N_HIDDEN + col] = acc[t][r] * sc + sh;
        }
    }
}

// ------------------------------------------- edge gather + scatter-add (atomics)
// 16 threads per edge, each handles 4 contiguous features (float4 gather).
__global__ void gnn_edge_scatter_kernel(const int* __restrict__ src,
                                        const int* __restrict__ dst,
                                        const float* __restrict__ h,
                                        float* __restrict__ agg) {
    const long t = (long)blockIdx.x * blockDim.x + threadIdx.x;
    if (t >= (long)N_EDGES * 16) return;
    const int e  = (int)(t >> 4);
    const int g  = (int)(t & 15);
    const int c0 = g * 4;
    const int s = src[e];
    const int d = dst[e];
    const float4 v = *(const float4*)(h + (long)s * N_HIDDEN + c0);
    float* ap = agg + (long)d * N_HIDDEN + c0;
    atomicAdd(ap + 0, v.x);
    atomicAdd(ap + 1, v.y);
    atomicAdd(ap + 2, v.z);
    atomicAdd(ap + 3, v.w);
}

// --------------------------- PReLU + global sum-pool (run-length compressed)
// seg_ids are sorted: each thread walks 64 consecutive nodes for one column,
// accumulating in a register and flushing via atomic only on segment change.
__global__ void gnn_prelu_pool_kernel(const float* __restrict__ agg,
                                      const int* __restrict__ seg,
                                      const float* __restrict__ alpha,
                                      float* __restrict__ pooled) {
    const int col = threadIdx.x;                       // 0..63
    const long chunk = (long)blockIdx.x * blockDim.y + threadIdx.y;
    long n0 = chunk * 64;
    if (n0 >= N_NODES) return;
    long n1 = n0 + 64;
    if (n1 > N_NODES) n1 = N_NODES;

    const float al = alpha[col];
    int cur = seg[n0];
    float sum = 0.0f;
    for (long n = n0; n < n1; ++n) {
        const int sg = seg[n];
        const float a = agg[n * N_HIDDEN + col];
        const float v = (a > 0.0f) ? a : al * a;
        if (sg != cur) {
            atomicAdd(&pooled[(long)cur * N_HIDDEN + col], sum);
            cur = sg;
            sum = 0.0f;
        }
        sum += v;
    }
    atomicAdd(&pooled[(long)cur * N_HIDDEN + col], sum);
}

// ---------------------------------------------------------------- Dense(1) head
__global__ void gnn_head_kernel(const float* __restrict__ pooled,
                                const float* __restrict__ Wd,
                                const float* __restrict__ bd,
                                float* __restrict__ out) {
    const int g = threadIdx.x;
    if (g < N_GRAPHS) {
        float s = bd[0];
        #pragma unroll
        for (int c = 0; c < N_HIDDEN; ++c) s += pooled[g * N_HIDDEN + c] * Wd[c];
        out[g] = s;
    }
}

extern "C" void kernel_launch(void* const* d_in, const int* in_sizes, int n_in,
                              void* d_out, int out_size, void* d_ws, size_t ws_size,
                              hipStream_t stream) {
    (void)in_sizes; (void)n_in; (void)out_size; (void)ws_size;
    const float* x        = (const float*)d_in[0];
    const int*   src      = (const int*)  d_in[1];
    const int*   dst      = (const int*)  d_in[2];
    const int*   seg_ids  = (const int*)  d_in[3];
    const float* W        = (const float*)d_in[4];
    const float* b        = (const float*)d_in[5];
    const float* gamma    = (const float*)d_in[6];
    const float* beta     = (const float*)d_in[7];
    const float* mov_mean = (const float*)d_in[8];
    const float* mov_var  = (const float*)d_in[9];
    const float* alpha    = (const float*)d_in[10];
    const float* Wd       = (const float*)d_in[11];
    const float* bd       = (const float*)d_in[12];
    float* out = (float*)d_out;

    // workspace layout (needs ~51.3 MB)
    float* h      = (float*)d_ws;                        // [N_NODES * 64]
    float* agg    = h + (long)N_NODES * N_HIDDEN;        // [N_NODES * 64]
    float* pooled = agg + (long)N_NODES * N_HIDDEN;      // [N_GRAPHS * 64]

    // 1) zero agg + pooled (contiguous)
    const long nzero = (long)N_NODES * N_HIDDEN + (long)N_GRAPHS * N_HIDDEN;
    gnn_zero_kernel<<<4096, 256, 0, stream>>>(agg, nzero);

    // 2) fused WMMA GEMM + BN -> h
    const int n_row_tiles = (N_NODES + 15) / 16;         // 6250 waves
    const int gemm_blocks = (n_row_tiles + 3) / 4;       // 4 waves / block
    gnn_gemm_bn_kernel<<<gemm_blocks, 128, 0, stream>>>(
        x, W, b, gamma, beta, mov_mean, mov_var, h);

    // 3) gather + scatter-add over edges
    const long edge_threads = (long)N_EDGES * 16;
    const int edge_blocks = (int)((edge_threads + 255) / 256);
    gnn_edge_scatter_kernel<<<edge_blocks, 256, 0, stream>>>(src, dst, h, agg);

    // 4) PReLU + pooled segment sum
    const int n_chunks = (N_NODES + 63) / 64;            // 1563
    dim3 pb(64, 4);
    const int pool_blocks = (n_chunks + 3) / 4;
    gnn_prelu_pool_kernel<<<pool_blocks, pb, 0, stream>>>(agg, seg_ids, alpha, pooled);

    // 5) dense head
    gnn_head_kernel<<<1, 128, 0, stream>>>(pooled, Wd, bd, out);
}